// MACEDiffusionAdapted_8452495639066
// MI455X (gfx1250) — compile-verified
//
#include <hip/hip_runtime.h>
#include <math.h>

typedef __attribute__((ext_vector_type(2))) float v2f;
typedef __attribute__((ext_vector_type(8))) float v8f;
typedef __attribute__((ext_vector_type(4))) unsigned int v4u;
typedef __attribute__((ext_vector_type(8))) int v8i;
typedef __attribute__((ext_vector_type(4))) int v4i;

#define N_NODES 512
#define F_DIM   64
#define NF      (N_NODES * F_DIM)
#define AVG_INV (1.0f / 511.0f)

__device__ __forceinline__ float silu_f(float x) { return x / (1.0f + expf(-x)); }

__device__ __forceinline__ v8f wmma_f32_16x16x4(v2f a, v2f b, v8f c) {
  // 8 args: (neg_a, A, neg_b, B, c_mod, C, reuse_a, reuse_b)
  return __builtin_amdgcn_wmma_f32_16x16x4_f32(false, a, false, b, (short)0, c, false, false);
}

// ---------------------------------------------------------------------------
// Init: h0 = embed(onehot(species) ++ time_embedding) @ W_embed + b_embed
//       h1 = 0 ; posw = positions
// ---------------------------------------------------------------------------
__global__ void k_init(const float* __restrict__ positions,
                       const int*   __restrict__ node_features,
                       const float* __restrict__ te,
                       const float* __restrict__ W_embed,
                       const float* __restrict__ b_embed,
                       float* __restrict__ h0, float* __restrict__ h1,
                       float* __restrict__ posw)
{
  int idx = blockIdx.x * blockDim.x + threadIdx.x;
  if (idx < NF) {
    int n = idx >> 6, f = idx & 63;
    int sp = node_features[n] - 1;
    float acc = b_embed[f] + W_embed[sp * F_DIM + f];
    for (int t = 0; t < 32; ++t)
      acc += te[t] * W_embed[(5 + t) * F_DIM + f];
    h0[idx] = acc;
    h1[0 * NF + idx] = 0.f;
    h1[1 * NF + idx] = 0.f;
    h1[2 * NF + idx] = 0.f;
  }
  if (idx < N_NODES * 3) posw[idx] = positions[idx];
}

// ---------------------------------------------------------------------------
// Generic f32 WMMA GEMM: C[M,N] = act(A[M,K] @ B[K,N] + bias)
// grid = (M/16, N/16), block = 32 (one wave per 16x16 tile)
// ---------------------------------------------------------------------------
__global__ void k_gemm16(const float* __restrict__ A, const float* __restrict__ B,
                         const float* __restrict__ bias, float* __restrict__ C,
                         int K, int Ncols, int act)
{
  int lane = threadIdx.x;
  int half = lane >> 4, lm = lane & 15;
  int mrow = blockIdx.x * 16;
  int ncol = blockIdx.y * 16;
  v8f acc = {0.f, 0.f, 0.f, 0.f, 0.f, 0.f, 0.f, 0.f};
  for (int k0 = 0; k0 < K; k0 += 4) {
    v2f a, b;
    // A 16x4: lanes 0-15 -> K = k0+{0,1}; lanes 16-31 -> K = k0+{2,3}
    const float* ap = A + (size_t)(mrow + lm) * K + k0 + 2 * half;
    a.x = ap[0]; a.y = ap[1];
    // B 4x16: VGPR0 = rows {0,2}, VGPR1 = rows {1,3}, lanes = N
    const float* bp = B + (size_t)(k0 + 2 * half) * Ncols + ncol + lm;
    b.x = bp[0]; b.y = bp[Ncols];
    acc = wmma_f32_16x16x4(a, b, acc);
  }
  float bs = bias ? bias[ncol + lm] : 0.f;
  for (int v = 0; v < 8; ++v) {
    int m = v + half * 8;             // D: VGPR v holds M=v (lanes<16) / v+8
    float val = acc[v] + bs;
    if (act) val = silu_f(val);
    C[(size_t)(mrow + m) * Ncols + ncol + lm] = val;
  }
}

// ---------------------------------------------------------------------------
// Fused edge kernel: one workgroup per receiver r (complete graph).
// W_e2 [32,384] staged global->LDS by the Tensor Data Mover (one
// tensor_load_to_lds per workgroup, completed with s_wait_tensorcnt).
// Chunks of 16 senders: build he[16x32] & Y[16x9] in LDS, compute
// w = he @ W_e2 ([16x384]) with WMMA f32 16x16x4, scatter messages into
// LDS accumulators A0[F], A1[F,3], A2[F,5] with ds_add_f32.
// ---------------------------------------------------------------------------
__global__ void __launch_bounds__(128) k_edge(
    const float* __restrict__ posw,
    const float* __restrict__ x0,      // [N,F]
    const float* __restrict__ x1,      // 3 planes of [N,F]
    const float* __restrict__ We1,     // [32]
    const float* __restrict__ be1,     // [32]
    const float* __restrict__ We2,     // [32,384]
    float* __restrict__ A0g, float* __restrict__ A1g, float* __restrict__ A2g)
{
  __shared__ float sW[32 * 384];       // 48 KB, TDM-loaded once per workgroup
  __shared__ float sY[16][12];         // [0..8]=Y, [9]=rr, [10]=valid
  __shared__ float sHe[16][32];
  __shared__ float sA0[F_DIM];
  __shared__ float sA1[F_DIM * 3];
  __shared__ float sA2[F_DIM * 5];

  int r   = blockIdx.x;
  int tid = threadIdx.x;

  // ---- TDM: DMA W_e2 (2D tensor, 32 lines x 384 f32) into LDS ----
  if (tid < 32) {
    unsigned lds_off = (unsigned)(size_t)(void*)sW;
    unsigned long long ga = (unsigned long long)(size_t)We2;
    // D# group0: count=1 | lds_addr[63:32] | global_addr[120:64] | type=2[127:126]
    v4u g0 = { 1u,
               lds_off,
               (unsigned)(ga & 0xFFFFFFFFull),
               (unsigned)((ga >> 32) & 0x1FFFFFFull) | 0x80000000u };
    // D# group1: data_size=2(4B)[17:16]; tensor_dim0=384[79:48];
    //            tensor_dim1=32[111:80]; tile_dim0=384[127:112];
    //            tile_dim1=32[143:128]; tensor_dim0_stride=384[207:160]
    v8i g1 = { 0x00020000,
               (int)(384u << 16),
               (int)(32u << 16),
               (int)(384u << 16),
               32,
               384,
               0,
               0 };
    v4i gz4 = { 0, 0, 0, 0 };          // groups 2/3 unused (tile_dim2/3 = 0)
    v8i gz8 = { 0, 0, 0, 0, 0, 0, 0, 0 };
    __builtin_amdgcn_tensor_load_to_lds(g0, g1, gz4, gz4, gz8, 0);
    __builtin_amdgcn_s_wait_tensorcnt(0);
  }
  for (int i = tid; i < F_DIM;     i += 128) sA0[i] = 0.f;
  for (int i = tid; i < F_DIM * 3; i += 128) sA1[i] = 0.f;
  for (int i = tid; i < F_DIM * 5; i += 128) sA2[i] = 0.f;
  float prx = posw[r * 3 + 0], pry = posw[r * 3 + 1], prz = posw[r * 3 + 2];
  __syncthreads();

  int wave = tid >> 5, lane = tid & 31;
  int half = lane >> 4, lm = lane & 15;

  for (int cb = 0; cb < N_NODES; cb += 16) {
    // ---- phase 1a: per-edge geometry (Y, rr) for 16 senders ----
    if (tid < 16) {
      int s = cb + tid;
      if (s == r) {                    // no self-edge: zero -> zero contribution
        for (int q = 0; q < 11; ++q) sY[tid][q] = 0.f;
      } else {
        float vx = prx - posw[s * 3 + 0];
        float vy = pry - posw[s * 3 + 1];
        float vz = prz - posw[s * 3 + 2];
        float rr  = sqrtf(vx * vx + vy * vy + vz * vz);
        float inv = 1.f / rr;
        float ux = vx * inv, uy = vy * inv, uz = vz * inv;
        const float s3c  = 1.7320508075688772f;   // sqrt(3)
        const float s5c  = 2.2360679774997896f;   // sqrt(5)
        const float s15c = 3.8729833462074170f;   // sqrt(15)
        sY[tid][0] = 1.f;
        sY[tid][1] = s3c * ux; sY[tid][2] = s3c * uy; sY[tid][3] = s3c * uz;
        sY[tid][4] = s15c * ux * uy;
        sY[tid][5] = s15c * uy * uz;
        sY[tid][6] = 0.5f * s5c * (3.f * uz * uz - 1.f);
        sY[tid][7] = s15c * ux * uz;
        sY[tid][8] = 0.5f * s15c * (ux * ux - uy * uy);
        sY[tid][9]  = rr;
        sY[tid][10] = 1.f;             // valid flag
      }
    }
    __syncthreads();

    // ---- phase 1b: he[16x32] across all 128 threads ----
    for (int i = tid; i < 16 * 32; i += 128) {
      int e = i >> 5, k = i & 31;
      float rr    = sY[e][9];
      float valid = sY[e][10];
      sHe[e][k] = valid * silu_f(rr * We1[k] + be1[k]);
    }
    __syncthreads();

    // ---- phase 2+3: WMMA w-tile + message scatter (6 col-tiles / wave) ----
    for (int t = 0; t < 6; ++t) {
      int ct = wave + t * 4;           // col tiles 0..23 (384 cols)
      v8f acc = {0.f, 0.f, 0.f, 0.f, 0.f, 0.f, 0.f, 0.f};
      for (int k0 = 0; k0 < 32; k0 += 4) {
        v2f a, b;
        a.x = sHe[lm][k0 + 2 * half];
        a.y = sHe[lm][k0 + 2 * half + 1];
        b.x = sW[(k0 + 2 * half)     * 384 + ct * 16 + lm];
        b.y = sW[(k0 + 2 * half + 1) * 384 + ct * 16 + lm];
        acc = wmma_f32_16x16x4(a, b, acc);
      }
      int col = ct * 16 + lm;
      int f = col / 6, j = col - f * 6;
      for (int v = 0; v < 8; ++v) {
        int m = v + half * 8;          // edge-in-chunk for this accum element
        float wv = acc[v];
        int s = cb + m;
        float s0v = x0[s * F_DIM + f];
        if (j == 0) {
          atomicAdd(&sA0[f], wv * s0v);
        } else if (j == 1) {
          atomicAdd(&sA1[f * 3 + 0], wv * s0v * sY[m][1]);
          atomicAdd(&sA1[f * 3 + 1], wv * s0v * sY[m][2]);
          atomicAdd(&sA1[f * 3 + 2], wv * s0v * sY[m][3]);
        } else if (j == 2) {
          for (int q = 0; q < 5; ++q)
            atomicAdd(&sA2[f * 5 + q], wv * s0v * sY[m][4 + q]);
        } else if (j == 3) {
          float d = x1[0 * NF + s * F_DIM + f] * sY[m][1]
                  + x1[1 * NF + s * F_DIM + f] * sY[m][2]
                  + x1[2 * NF + s * F_DIM + f] * sY[m][3];
          atomicAdd(&sA0[f], wv * d);
        } else if (j == 4) {
          float sx = x1[0 * NF + s * F_DIM + f];
          float sy = x1[1 * NF + s * F_DIM + f];
          float sz = x1[2 * NF + s * F_DIM + f];
          float yx = sY[m][1], yy = sY[m][2], yz = sY[m][3];
          atomicAdd(&sA1[f * 3 + 0], wv * (sy * yz - sz * yy));
          atomicAdd(&sA1[f * 3 + 1], wv * (sz * yx - sx * yz));
          atomicAdd(&sA1[f * 3 + 2], wv * (sx * yy - sy * yx));
        }
        // j == 5: s2 == 0 always (channels 4:9 never populated) -> no term
      }
    }
    __syncthreads();
  }

  // ---- write accumulators (each r owned by exactly one workgroup) ----
  for (int i = tid; i < F_DIM; i += 128) A0g[r * F_DIM + i] = sA0[i];
  for (int i = tid; i < F_DIM * 3; i += 128) {
    int c = i / F_DIM, f = i - c * F_DIM;
    A1g[c * NF + r * F_DIM + f] = sA1[f * 3 + c];
  }
  for (int i = tid; i < F_DIM * 5; i += 128) {
    int c = i / F_DIM, f = i - c * F_DIM;
    A2g[c * NF + r * F_DIM + f] = sA2[f * 5 + c];
  }
}

// ---------------------------------------------------------------------------
// Per-(n,f) nonlinear node update -> h0 (out0), h1 (out1 planes)
// ---------------------------------------------------------------------------
__global__ void k_node(const float* __restrict__ A0g, const float* __restrict__ A1g,
                       const float* __restrict__ A2g, const int* __restrict__ node_features,
                       const float* __restrict__ Wp0, const float* __restrict__ Wp1,
                       float* __restrict__ h0, float* __restrict__ h1)
{
  int idx = blockIdx.x * blockDim.x + threadIdx.x;
  if (idx >= NF) return;
  int n = idx >> 6, f = idx & 63;
  float a0  = A0g[idx] * AVG_INV;
  float a1x = A1g[0 * NF + idx] * AVG_INV;
  float a1y = A1g[1 * NF + idx] * AVG_INV;
  float a1z = A1g[2 * NF + idx] * AVG_INV;
  float n1 = a1x * a1x + a1y * a1y + a1z * a1z;
  float n2 = 0.f;
  for (int q = 0; q < 5; ++q) {
    float a2 = A2g[q * NF + idx] * AVG_INV;
    n2 += a2 * a2;
  }
  int sp = node_features[n] - 1;
  const float* w0 = Wp0 + (size_t)(sp * F_DIM + f) * 6;
  float a02 = a0 * a0;
  float out0 = w0[0] * a0 + w0[1] * a02 + w0[2] * a02 * a0
             + w0[3] * n1 + w0[4] * n2 + w0[5] * a0 * n1;
  const float* w1 = Wp1 + (size_t)(sp * F_DIM + f) * 3;
  float g = w1[0] + w1[1] * a0 + w1[2] * a02;
  h0[idx] = out0;
  h1[0 * NF + idx] = g * a1x;
  h1[1 * NF + idx] = g * a1y;
  h1[2 * NF + idx] = g * a1z;
}

// ---------------------------------------------------------------------------
// pos += einsum('nf,nfc->nc', gate, out1)
// ---------------------------------------------------------------------------
__global__ void k_pos(const float* __restrict__ gate, const float* __restrict__ h1,
                      float* __restrict__ posw)
{
  int n = blockIdx.x * blockDim.x + threadIdx.x;
  if (n >= N_NODES) return;
  float vx = 0.f, vy = 0.f, vz = 0.f;
  for (int f = 0; f < F_DIM; ++f) {
    float gv = gate[n * F_DIM + f];
    vx += gv * h1[0 * NF + n * F_DIM + f];
    vy += gv * h1[1 * NF + n * F_DIM + f];
    vz += gv * h1[2 * NF + n * F_DIM + f];
  }
  posw[n * 3 + 0] += vx;
  posw[n * 3 + 1] += vy;
  posw[n * 3 + 2] += vz;
}

__global__ void k_out(const float* __restrict__ posw, const float* __restrict__ positions,
                      float* __restrict__ out)
{
  int idx = blockIdx.x * blockDim.x + threadIdx.x;
  if (idx < N_NODES * 3) out[idx] = posw[idx] - positions[idx];
}

// ---------------------------------------------------------------------------
extern "C" void kernel_launch(void* const* d_in, const int* in_sizes, int n_in,
                              void* d_out, int out_size, void* d_ws, size_t ws_size,
                              hipStream_t stream)
{
  (void)in_sizes; (void)n_in; (void)out_size; (void)ws_size;
  const float* positions     = (const float*)d_in[0];
  const int*   node_features = (const int*)  d_in[1];
  const float* te            = (const float*)d_in[2];
  // d_in[3]/d_in[4] (senders/receivers) are implied by the complete graph
  const float* W_embed = (const float*)d_in[5];
  const float* b_embed = (const float*)d_in[6];
  const float* W_lin   = (const float*)d_in[7];
  const float* W_e1    = (const float*)d_in[8];
  const float* b_e1    = (const float*)d_in[9];
  const float* W_e2    = (const float*)d_in[10];
  const float* Wp0     = (const float*)d_in[11];
  const float* Wp1     = (const float*)d_in[12];
  const float* Wr1     = (const float*)d_in[13];
  const float* br1     = (const float*)d_in[14];
  const float* Wr2g    = (const float*)d_in[16];   // Wr2s (d_in[15]) unused

  float* ws   = (float*)d_ws;
  float* h0   = ws; ws += NF;
  float* h1   = ws; ws += 3 * NF;
  float* x0   = ws; ws += NF;
  float* x1   = ws; ws += 3 * NF;
  float* A0   = ws; ws += NF;
  float* A1   = ws; ws += 3 * NF;
  float* A2   = ws; ws += 5 * NF;
  float* hr   = ws; ws += NF;
  float* gate = ws; ws += NF;
  float* posw = ws; ws += N_NODES * 3;

  k_init<<<(NF + 255) / 256, 256, 0, stream>>>(positions, node_features, te,
                                               W_embed, b_embed, h0, h1, posw);

  dim3 gg(N_NODES / 16, F_DIM / 16);   // 32 x 4 tiles, one wave each
  for (int l = 0; l < 2; ++l) {
    const float* Wl0 = W_lin + (size_t)(l * 3 + 0) * F_DIM * F_DIM;
    const float* Wl1 = W_lin + (size_t)(l * 3 + 1) * F_DIM * F_DIM;
    k_gemm16<<<gg, 32, 0, stream>>>(h0, Wl0, nullptr, x0, F_DIM, F_DIM, 0);
    for (int c = 0; c < 3; ++c)
      k_gemm16<<<gg, 32, 0, stream>>>(h1 + c * NF, Wl1, nullptr, x1 + c * NF,
                                      F_DIM, F_DIM, 0);
    k_edge<<<N_NODES, 128, 0, stream>>>(posw, x0, x1,
                                        W_e1 + l * 32, b_e1 + l * 32,
                                        W_e2 + (size_t)l * 32 * 384, A0, A1, A2);
    k_node<<<(NF + 255) / 256, 256, 0, stream>>>(A0, A1, A2, node_features,
                                                 Wp0 + (size_t)l * 5 * F_DIM * 6,
                                                 Wp1 + (size_t)l * 5 * F_DIM * 3,
                                                 h0, h1);
    k_gemm16<<<gg, 32, 0, stream>>>(h0, Wr1 + (size_t)l * F_DIM * 64,
                                    br1 + l * 64, hr, F_DIM, 64, 1);
    k_gemm16<<<gg, 32, 0, stream>>>(hr, Wr2g + (size_t)l * 64 * F_DIM,
                                    nullptr, gate, 64, F_DIM, 0);
    k_pos<<<2, 256, 0, stream>>>(gate, h1, posw);
  }
  k_out<<<6, 256, 0, stream>>>(posw, positions, (float*)d_out);
}